// CustomSoftMax_56453050139112
// MI455X (gfx1250) — compile-verified
//
#include <hip/hip_runtime.h>
#include <cstdint>

typedef float v4f __attribute__((ext_vector_type(4)));
typedef int   v4i __attribute__((ext_vector_type(4)));

#if defined(__HIP_DEVICE_COMPILE__)
typedef __attribute__((address_space(1))) v4i* gv4i_p;   // global int4*
typedef __attribute__((address_space(3))) v4i* lv4i_p;   // LDS int4*
#endif

#define D_DIM   128
#define OH      120
#define NSEC    6
#define SEC     20
#define RPB     32                      // rows per block
#define THREADS 256
#define ELTS_PER_BLOCK (RPB * D_DIM)    // 4096 floats = 16 KB LDS tile

#if defined(__HIP_DEVICE_COMPILE__)
#  if __has_builtin(__builtin_amdgcn_global_load_async_to_lds_b128)
#    define USE_ASYNC_BUILTIN 1
#  else
#    define USE_ASYNC_ASM 1
#  endif
#  if __has_builtin(__builtin_amdgcn_s_wait_asynccnt)
#    define WAIT_ASYNC() __builtin_amdgcn_s_wait_asynccnt(0)
#  else
#    define WAIT_ASYNC() asm volatile("s_wait_asynccnt 0x0" ::: "memory")
#  endif
#else
#  define WAIT_ASYNC() ((void)0)
#endif

__global__ __launch_bounds__(THREADS) void
CustomSoftMax_sections_kernel(const float* __restrict__ in,
                              float* __restrict__ out,
                              int nrows)
{
    __shared__ __align__(16) float lds[ELTS_PER_BLOCK];

    const int       t   = threadIdx.x;
    const long long r0  = (long long)blockIdx.x * RPB;        // first row of tile
    const bool full_tile = (r0 + RPB) <= (long long)nrows;

    // ---------------- Phase 1: stage 32x128 tile into LDS ----------------
    if (full_tile) {
        const float* gbase = in + r0 * D_DIM;                 // contiguous 16 KB
#if USE_ASYNC_BUILTIN
        #pragma unroll
        for (int i = 0; i < 4; ++i) {
            const int elt = t * 4 + i * 1024;                 // float index in tile
            __builtin_amdgcn_global_load_async_to_lds_b128(
                (gv4i_p)(gbase + elt),
                (lv4i_p)(&lds[elt]),
                0, 0);
        }
        WAIT_ASYNC();
#elif USE_ASYNC_ASM
        const unsigned lds_base = (unsigned)(uintptr_t)(&lds[0]);
        #pragma unroll
        for (int i = 0; i < 4; ++i) {
            const int elt = t * 4 + i * 1024;
            const unsigned long long gaddr =
                (unsigned long long)(uintptr_t)(gbase + elt);
            const unsigned laddr = lds_base + (unsigned)(elt * 4);
            asm volatile("global_load_async_to_lds_b128 %0, %1, off"
                         :: "v"(laddr), "v"(gaddr) : "memory");
        }
        WAIT_ASYNC();
#else
        // host-compile parse path / non-gfx fallback
        #pragma unroll
        for (int i = 0; i < 4; ++i) {
            const int elt = t * 4 + i * 1024;
            *(v4f*)&lds[elt] = *(const v4f*)(gbase + elt);
        }
#endif
    } else {
        // partial tile (not hit for B % 32 == 0): guarded scalar fill
        for (int e = t; e < ELTS_PER_BLOCK; e += THREADS) {
            const long long r = r0 + e / D_DIM;
            lds[e] = (r < (long long)nrows) ? in[r * D_DIM + (e % D_DIM)] : 0.0f;
        }
    }
    __syncthreads();

    // ---------------- Phase 2: per-section softmax ----------------
    if (t < RPB * NSEC) {                       // 192 section threads
        const int row  = t / NSEC;
        const int sec  = t - row * NSEC;
        const long long grow = r0 + row;
        if (grow < (long long)nrows) {
            const float* lsec = &lds[row * D_DIM + sec * SEC]; // 16B aligned
            float x[SEC];
            #pragma unroll
            for (int i = 0; i < 5; ++i)
                *(v4f*)&x[i * 4] = *(const v4f*)(lsec + i * 4); // ds_load_b128

            float m = x[0];
            #pragma unroll
            for (int i = 1; i < SEC; ++i) m = fmaxf(m, x[i]);

            float s = 0.0f;
            #pragma unroll
            for (int i = 0; i < SEC; ++i) { x[i] = __expf(x[i] - m); s += x[i]; }

#if defined(__HIP_DEVICE_COMPILE__)
            const float r = __builtin_amdgcn_rcpf(s);
#else
            const float r = 1.0f / s;
#endif
            float* osec = out + grow * D_DIM + sec * SEC;      // 16B aligned
            #pragma unroll
            for (int i = 0; i < 5; ++i) {
                v4f v = { x[i*4+0] * r, x[i*4+1] * r, x[i*4+2] * r, x[i*4+3] * r };
                __builtin_nontemporal_store(v, (v4f*)(osec + i * 4));
            }
        }
    } else {                                     // 64 tail threads: zero cols 120..127
        const int u    = t - RPB * NSEC;         // 0..63
        const int row  = u >> 1;
        const int half = u & 1;
        const long long grow = r0 + row;
        if (grow < (long long)nrows) {
            v4f z = {0.0f, 0.0f, 0.0f, 0.0f};
            __builtin_nontemporal_store(z, (v4f*)(out + grow * D_DIM + OH + half * 4));
        }
    }
}

extern "C" void kernel_launch(void* const* d_in, const int* in_sizes, int n_in,
                              void* d_out, int out_size, void* d_ws, size_t ws_size,
                              hipStream_t stream)
{
    (void)n_in; (void)out_size; (void)d_ws; (void)ws_size;
    const float* in  = (const float*)d_in[0];
    float*       out = (float*)d_out;
    const int nrows  = in_sizes[0] / D_DIM;            // 1,000,000
    const int blocks = (nrows + RPB - 1) / RPB;        // 31,250
    CustomSoftMax_sections_kernel<<<blocks, THREADS, 0, stream>>>(in, out, nrows);
}